// GraphEmbedder_73400991088674
// MI455X (gfx1250) — compile-verified
//
#include <hip/hip_runtime.h>

typedef __attribute__((ext_vector_type(2))) float v2f;
typedef __attribute__((ext_vector_type(8))) float v8f;

// ---------------------------------------------------------------------------
// FP32 WMMA GEMM: C[M,NCOL] = A[M,K] * B[K,NCOL]
// One wave computes one 16x16 output tile via V_WMMA_F32_16X16X4_F32.
// A frag (16x4 f32): lane l holds A[l%16][k0], A[l%16][k0+1], k0 = 2*(l/16)
// B frag (4x16 f32): lane l holds B[k0][l%16], B[k0+1][l%16]
// C/D (16x16 f32):   VGPR r, lane l -> row r + 8*(l/16), col l%16
// ---------------------------------------------------------------------------
template <int K, int NCOL>
__global__ __launch_bounds__(256) void gemm_wmma(const float* __restrict__ A,
                                                 const float* __restrict__ B,
                                                 float* __restrict__ C, int M) {
    const int colTiles = NCOL / 16;
    const int gtid = blockIdx.x * 256 + threadIdx.x;
    const int wave = gtid >> 5;
    const int lane = threadIdx.x & 31;
    const int tm = wave / colTiles;
    const int tn = wave % colTiles;
    if (tm * 16 >= M) return;  // wave-uniform: EXEC stays all-ones for WMMA

    const int mr = lane & 15;
    const int hi = lane >> 4;     // 0 or 1
    const int khalf = hi * 2;     // 0 or 2
    const int nc = tn * 16 + mr;  // output column for this lane

    const float* __restrict__ Arow = A + (size_t)(tm * 16 + mr) * K;
    v8f acc = {};
#pragma unroll 4
    for (int k = 0; k < K; k += 4) {
        v2f a = *reinterpret_cast<const v2f*>(Arow + k + khalf);
        v2f b;
        b.x = B[(size_t)(k + khalf) * NCOL + nc];
        b.y = B[(size_t)(k + khalf + 1) * NCOL + nc];
        acc = __builtin_amdgcn_wmma_f32_16x16x4_f32(
            /*neg_a=*/false, a, /*neg_b=*/false, b,
            /*c_mod=*/(short)0, acc, /*reuse_a=*/false, /*reuse_b=*/false);
    }
    const int rbase = tm * 16 + hi * 8;
#pragma unroll
    for (int r = 0; r < 8; ++r)
        C[(size_t)(rbase + r) * NCOL + nc] = acc[r];
}

// ---------------------------------------------------------------------------
// Attention projections: a_src[n,h] = dot(h[n,h,:], att_src[h,:]) (same for dst)
// ---------------------------------------------------------------------------
__global__ void att_proj(const float* __restrict__ h, const float* __restrict__ att_s,
                         const float* __restrict__ att_d, float* __restrict__ a_s,
                         float* __restrict__ a_d, long long N, int H, int C) {
    long long i = blockIdx.x * (long long)blockDim.x + threadIdx.x;
    if (i >= N * H) return;
    long long n = i / H;
    int hh = (int)(i % H);
    const float* row = h + (size_t)n * H * C + (size_t)hh * C;
    float ss = 0.f, dd = 0.f;
    for (int c = 0; c < C; ++c) {
        float v = row[c];
        ss += v * att_s[hh * C + c];
        dd += v * att_d[hh * C + c];
    }
    a_s[i] = ss;
    a_d[i] = dd;
}

// Monotone float <-> uint mapping for atomicMax on signed floats.
__device__ __forceinline__ unsigned f2ord(float f) {
    unsigned u = __float_as_uint(f);
    return (u & 0x80000000u) ? ~u : (u | 0x80000000u);
}
__device__ __forceinline__ float ord2f(unsigned u) {
    return (u & 0x80000000u) ? __uint_as_float(u & 0x7FFFFFFFu) : __uint_as_float(~u);
}

__device__ __forceinline__ void edge_sd(const long long* __restrict__ ei, long long i,
                                        long long E, long long& s, long long& d) {
    if (i < E) { s = ei[i]; d = ei[E + i]; }
    else       { s = d = i - E; }  // appended self-loops
}

// Pass 1: e = leaky_relu(a_src[s] + a_dst[d]); segment max over dst (atomic).
template <int H>
__global__ void edge_logits(const long long* __restrict__ ei, long long E, long long Et,
                            const float* __restrict__ a_s, const float* __restrict__ a_d,
                            float* __restrict__ e, unsigned* __restrict__ emax) {
    long long i = blockIdx.x * (long long)blockDim.x + threadIdx.x;
    if (i >= Et) return;
    long long s, d;
    edge_sd(ei, i, E, s, d);
#pragma unroll
    for (int h = 0; h < H; ++h) {
        float v = a_s[s * H + h] + a_d[d * H + h];
        v = (v > 0.f) ? v : 0.2f * v;  // leaky_relu, PyG default slope
        e[i * H + h] = v;
        atomicMax(&emax[d * H + h], f2ord(v));
    }
}

// Pass 2: ex = exp(e - max[dst]); segment sum over dst (atomic); e <- ex in place.
template <int H>
__global__ void edge_exp(const long long* __restrict__ ei, long long E, long long Et,
                         float* __restrict__ e, const unsigned* __restrict__ emax,
                         float* __restrict__ denom) {
    long long i = blockIdx.x * (long long)blockDim.x + threadIdx.x;
    if (i >= Et) return;
    long long s, d;
    edge_sd(ei, i, E, s, d);
#pragma unroll
    for (int h = 0; h < H; ++h) {
        float ex = __expf(e[i * H + h] - ord2f(emax[d * H + h]));
        e[i * H + h] = ex;
        atomicAdd(&denom[d * H + h], ex);
    }
}

// Pass 3: agg[dst, :] += h[src, :] * (ex / denom[dst]); one edge per block,
// lane-per-channel -> coalesced reads of h[src] and coalesced f32 atomics.
template <int H, int C>
__global__ void edge_aggr(const long long* __restrict__ ei, long long E,
                          const float* __restrict__ hfeat, const float* __restrict__ ex,
                          const float* __restrict__ denom, float* __restrict__ agg) {
    long long i = blockIdx.x;  // edge id (grid.x == Et)
    int c = threadIdx.x;       // 0 .. H*C-1
    int hh = c / C;
    long long s, d;
    edge_sd(ei, i, E, s, d);
    float alpha = ex[i * H + hh] / (denom[d * H + hh] + 1e-16f);
    atomicAdd(&agg[d * (H * C) + c], hfeat[s * (H * C) + c] * alpha);
}

// Layer-1 epilogue: head mean + bias + ELU
__global__ void final1(const float* __restrict__ agg, const float* __restrict__ bias,
                       float* __restrict__ x1, long long N) {
    long long i = blockIdx.x * (long long)blockDim.x + threadIdx.x;
    if (i >= N * 64) return;
    long long n = i >> 6;
    int c = (int)(i & 63);
    float v = 0.5f * (agg[(size_t)n * 128 + c] + agg[(size_t)n * 128 + 64 + c]) + bias[c];
    x1[i] = (v > 0.f) ? v : (__expf(v) - 1.0f);  // ELU(alpha=1)
}

// Layer-2 epilogue: bias add
__global__ void final2(const float* __restrict__ agg, const float* __restrict__ bias,
                       float* __restrict__ out, long long N) {
    long long i = blockIdx.x * (long long)blockDim.x + threadIdx.x;
    if (i >= N * 64) return;
    out[i] = agg[i] + bias[i & 63];
}

extern "C" void kernel_launch(void* const* d_in, const int* in_sizes, int n_in,
                              void* d_out, int out_size, void* d_ws, size_t ws_size,
                              hipStream_t stream) {
    const float*     x    = (const float*)d_in[0];
    const long long* ei   = (const long long*)d_in[1];  // int64 [2,E]
    const float*     W1   = (const float*)d_in[2];      // [128,128]
    const float*     as1w = (const float*)d_in[3];      // [2,64]
    const float*     ad1w = (const float*)d_in[4];
    const float*     b1   = (const float*)d_in[5];
    const float*     W2   = (const float*)d_in[6];      // [64,64]
    const float*     as2w = (const float*)d_in[7];      // [1,64]
    const float*     ad2w = (const float*)d_in[8];
    const float*     b2   = (const float*)d_in[9];

    const long long N  = in_sizes[0] / 128;
    const long long E  = in_sizes[1] / 2;
    const long long Et = E + N;  // with self-loops

    // Workspace layout (256B-aligned slabs)
    char* wsb = (char*)d_ws;
    size_t off = 0;
    auto alloc = [&](size_t bytes) -> void* {
        void* p = wsb + off;
        off = (off + bytes + 255) & ~(size_t)255;
        return p;
    };
    float*    h1     = (float*)alloc((size_t)N * 128 * 4);
    float*    a_s1   = (float*)alloc((size_t)N * 2 * 4);
    float*    a_d1   = (float*)alloc((size_t)N * 2 * 4);
    unsigned* emax1  = (unsigned*)alloc((size_t)N * 2 * 4);
    float*    denom1 = (float*)alloc((size_t)N * 2 * 4);
    float*    e1     = (float*)alloc((size_t)Et * 2 * 4);
    float*    agg1   = (float*)alloc((size_t)N * 128 * 4);
    float*    x1     = (float*)alloc((size_t)N * 64 * 4);
    float*    h2     = (float*)alloc((size_t)N * 64 * 4);
    float*    a_s2   = (float*)alloc((size_t)N * 4);
    float*    a_d2   = (float*)alloc((size_t)N * 4);
    unsigned* emax2  = (unsigned*)alloc((size_t)N * 4);
    float*    denom2 = (float*)alloc((size_t)N * 4);
    float*    e2     = (float*)alloc((size_t)Et * 4);
    float*    agg2   = (float*)alloc((size_t)N * 64 * 4);

    // Zero accumulators every call (graph-capturable memset nodes).
    hipMemsetAsync(emax1, 0, (size_t)N * 2 * 4, stream);   // ord(0)=0 < ord(any float written)
    hipMemsetAsync(denom1, 0, (size_t)N * 2 * 4, stream);
    hipMemsetAsync(agg1, 0, (size_t)N * 128 * 4, stream);
    hipMemsetAsync(emax2, 0, (size_t)N * 4, stream);
    hipMemsetAsync(denom2, 0, (size_t)N * 4, stream);
    hipMemsetAsync(agg2, 0, (size_t)N * 64 * 4, stream);

    const int Mtiles = (int)((N + 15) / 16);
    const int eb = (int)((Et + 255) / 256);
    const int nb64 = (int)((N * 64 + 255) / 256);

    // ---------------- Layer 1 (heads=2, C=64) ----------------
    {
        long long waves = (long long)Mtiles * 8;  // 128/16 col tiles
        int blocks = (int)((waves * 32 + 255) / 256);
        gemm_wmma<128, 128><<<blocks, 256, 0, stream>>>(x, W1, h1, (int)N);
    }
    {
        int blocks = (int)((N * 2 + 255) / 256);
        att_proj<<<blocks, 256, 0, stream>>>(h1, as1w, ad1w, a_s1, a_d1, N, 2, 64);
    }
    edge_logits<2><<<eb, 256, 0, stream>>>(ei, E, Et, a_s1, a_d1, e1, emax1);
    edge_exp<2><<<eb, 256, 0, stream>>>(ei, E, Et, e1, emax1, denom1);
    edge_aggr<2, 64><<<(unsigned)Et, 128, 0, stream>>>(ei, E, h1, e1, denom1, agg1);
    final1<<<nb64, 256, 0, stream>>>(agg1, b1, x1, N);

    // ---------------- Layer 2 (heads=1, C=64) ----------------
    {
        long long waves = (long long)Mtiles * 4;  // 64/16 col tiles
        int blocks = (int)((waves * 32 + 255) / 256);
        gemm_wmma<64, 64><<<blocks, 256, 0, stream>>>(x1, W2, h2, (int)N);
    }
    {
        int blocks = (int)((N + 255) / 256);
        att_proj<<<blocks, 256, 0, stream>>>(h2, as2w, ad2w, a_s2, a_d2, N, 1, 64);
    }
    edge_logits<1><<<eb, 256, 0, stream>>>(ei, E, Et, a_s2, a_d2, e2, emax2);
    edge_exp<1><<<eb, 256, 0, stream>>>(ei, E, Et, e2, emax2, denom2);
    edge_aggr<1, 64><<<(unsigned)Et, 64, 0, stream>>>(ei, E, h2, e2, denom2, agg2);
    final2<<<nb64, 256, 0, stream>>>(agg2, b2, (float*)d_out, N);
}